// CorrelationLayer1D_13580686590316
// MI455X (gfx1250) — compile-verified
//
#include <hip/hip_runtime.h>

typedef __attribute__((ext_vector_type(2))) float v2f;
typedef __attribute__((ext_vector_type(8))) float v8f;

#define B_    8
#define C_    128
#define H_    160
#define W_    320
#define D_    81          // 2*40/1 + 1
#define PAD_  80          // MAX_DISP*2
#define WG_W  64          // w columns handled per workgroup
#define LDS1_STRIDE 132   // dword stride for [w][c] tiles: conflict-free b64 frag loads
#define LDS2_W (WG_W + PAD_)   // 144
#define G_STRIDE 100      // dword stride for G staging (96 cols needed)

__global__ __launch_bounds__(256) void corr1d_wmma(const float* __restrict__ x1,
                                                   const float* __restrict__ x2,
                                                   float* __restrict__ out)
{
    // x1 tile [64 w][128 c] (later reused as G staging), x2 tile [144 w][128 c]
    __shared__ __align__(16) float lds1[WG_W   * LDS1_STRIDE];   // 33,792 B
    __shared__ __align__(16) float lds2[LDS2_W * LDS1_STRIDE];   // 76,032 B

    const int t  = threadIdx.x;
    const int g  = blockIdx.x;
    const int wg = g % (W_ / WG_W);
    const int h  = (g / (W_ / WG_W)) % H_;
    const int b  = g / ((W_ / WG_W) * H_);
    const int w0 = wg * WG_W;

    // ---- stage x1[b, :, h, w0:w0+64] transposed -> lds1[w][c]
    {
        const float* p = x1 + ((size_t)(b * C_) * H_ + h) * (size_t)W_ + w0;
        for (int idx = t; idx < WG_W * C_; idx += 256) {
            int wl = idx & (WG_W - 1);
            int c  = idx >> 6;
            lds1[wl * LDS1_STRIDE + c] = p[(size_t)c * (H_ * W_) + wl];
        }
    }
    // ---- stage x2[b, :, h, w0-80 : w0+64] transposed, zero outside image
    {
        const float* p = x2 + ((size_t)(b * C_) * H_ + h) * (size_t)W_;
        for (int idx = t; idx < LDS2_W * C_; idx += 256) {
            int c  = idx / LDS2_W;
            int wl = idx - c * LDS2_W;
            int w2 = w0 - PAD_ + wl;
            float v = 0.0f;
            if (w2 >= 0 && w2 < W_) v = p[(size_t)c * (H_ * W_) + w2];
            lds2[wl * LDS1_STRIDE + c] = v;
        }
    }
    __syncthreads();

    // ---- banded Gram tiles via V_WMMA_F32_16X16X4_F32
    const int lane = t & 31;
    const int widx = t >> 5;
    const int ti   = widx >> 1;        // row tile 0..3
    const int jb   = (widx & 1) * 3;   // col tiles jb..jb+2 of 0..5
    const int mn   = lane & 15;        // M for A / N for B
    const int kh   = lane >> 4;        // upper half handles K+2,K+3

    // A[m][k] = x1[c=k][w0+16*ti+m] ; B[k][n] = x2pad[c=k][16*(ti+j)+n]
    const float* aB = &lds1[(16 * ti        + mn) * LDS1_STRIDE + 2 * kh];
    const float* bB = &lds2[(16 * (ti + jb) + mn) * LDS1_STRIDE + 2 * kh];

    v8f acc0 = {}; v8f acc1 = {}; v8f acc2 = {};
#pragma unroll
    for (int k0 = 0; k0 < C_; k0 += 4) {
        v2f a  = *(const v2f*)(aB + k0);
        v2f b0 = *(const v2f*)(bB + k0);
        v2f b1 = *(const v2f*)(bB + 16 * LDS1_STRIDE + k0);
        v2f b2 = *(const v2f*)(bB + 32 * LDS1_STRIDE + k0);
        acc0 = __builtin_amdgcn_wmma_f32_16x16x4_f32(false, a, false, b0, (short)0, acc0, false, false);
        acc1 = __builtin_amdgcn_wmma_f32_16x16x4_f32(false, a, false, b1, (short)0, acc1, false, false);
        acc2 = __builtin_amdgcn_wmma_f32_16x16x4_f32(false, a, false, b2, (short)0, acc2, false, false);
    }

    __syncthreads();   // everyone done reading lds1 before we reuse it for G

    // ---- stage G: lds1[(16*ti + M)*100 + 16*j + N]
#pragma unroll
    for (int r = 0; r < 8; ++r) {
        int M = r + 8 * kh;                       // D-matrix VGPR layout
        float* gp = &lds1[(16 * ti + M) * G_STRIDE + mn];
        gp[16 * (jb + 0)] = acc0[r];
        gp[16 * (jb + 1)] = acc1[r];
        gp[16 * (jb + 2)] = acc2[r];
    }
    __syncthreads();

    // ---- coalesced store: out[b, d, h, w0+wl] = G[i][M][M + d]
    {
        int wl = t & 63;
        int gi = wl >> 4;
        int M  = wl & 15;
        const float* grow = &lds1[(16 * gi + M) * G_STRIDE + M];
        float* op = out + (((size_t)b * D_) * H_ + h) * (size_t)W_ + w0 + wl;
        for (int d = (t >> 6); d < D_; d += 4) {
            op[(size_t)d * (H_ * W_)] = grow[d];
        }
    }
}

extern "C" void kernel_launch(void* const* d_in, const int* in_sizes, int n_in,
                              void* d_out, int out_size, void* d_ws, size_t ws_size,
                              hipStream_t stream) {
    const float* x1 = (const float*)d_in[0];
    const float* x2 = (const float*)d_in[1];
    float* out = (float*)d_out;
    dim3 grid(B_ * H_ * (W_ / WG_W));   // 8*160*5 = 6400 workgroups
    corr1d_wmma<<<grid, dim3(256), 0, stream>>>(x1, x2, out);
}